// GraphAttConv_2405181686104
// MI455X (gfx1250) — compile-verified
//
#include <hip/hip_runtime.h>
#include <hip/hip_bf16.h>

typedef __attribute__((ext_vector_type(2))) float v2f;
typedef __attribute__((ext_vector_type(8))) float v8f;

#define F_IN   128
#define HEADS  8
#define F_PH   16
#define F_OUT  128            // HEADS * F_PH
#define NEG_SLOPE 0.2f

// ---------------------------------------------------------------------------
// K0: init out = 0, emax = -inf, denom = 0
// ---------------------------------------------------------------------------
__global__ void gat_init(float* __restrict__ out, float* __restrict__ emax,
                         float* __restrict__ denom, int nOut, int nNH) {
  int i = blockIdx.x * blockDim.x + threadIdx.x;
  if (i < nOut) out[i] = 0.0f;
  if (i < nNH) {
    emax[i]  = __uint_as_float(0xFF800000u);  // -inf
    denom[i] = 0.0f;
  }
}

// ---------------------------------------------------------------------------
// K1: h[n, head*16+o] = sum_f x[n,f] * W[head,f,o]    via V_WMMA_F32_16X16X4_F32
// One wave per 16-row stripe; 8 accumulator tiles (one per head), K=128 in
// 32 steps of 4. fp32 WMMA matches reference precision.
// ---------------------------------------------------------------------------
__global__ __launch_bounds__(256) void gat_proj_wmma(
    const float* __restrict__ x, const float* __restrict__ W,
    float* __restrict__ h, int nrows) {
  const int wave = threadIdx.x >> 5;                 // 0..7
  const int lane = threadIdx.x & 31;
  const long long row0 = ((long long)blockIdx.x * 8 + wave) * 16;
  if (row0 >= nrows) return;                         // wave-uniform exit

  const int m     = lane & 15;                       // row/col within tile
  const int khalf = (lane >> 4) << 1;                // 0 or 2
  long long rowA = row0 + m;
  if (rowA >= nrows) rowA = nrows - 1;               // clamp (EXEC stays full)
  const float* xrow = x + rowA * F_IN;

  v8f acc[HEADS];
  #pragma unroll
  for (int t = 0; t < HEADS; ++t) acc[t] = (v8f){0,0,0,0,0,0,0,0};

  for (int kt = 0; kt < F_IN / 4; ++kt) {
    const int kbase = kt * 4 + khalf;
    // A fragment: x[m][kbase], x[m][kbase+1]  (8B-aligned pair)
    v2f a = *(const v2f*)(xrow + kbase);
    #pragma unroll
    for (int t = 0; t < HEADS; ++t) {
      // B fragment: W[t][kbase][m], W[t][kbase+1][m]
      const float* wb = W + t * (F_IN * F_PH) + kbase * F_PH + m;
      v2f b;
      b.x = wb[0];
      b.y = wb[F_PH];
      acc[t] = __builtin_amdgcn_wmma_f32_16x16x4_f32(
          /*neg_a=*/false, a, /*neg_b=*/false, b,
          /*c_mod=*/(short)0, acc[t], /*reuse_a=*/false, /*reuse_b=*/false);
    }
  }

  // C/D layout: VGPR v -> M = v + (lane>=16 ? 8 : 0), N = lane&15
  const int mofs = (lane >> 4) << 3;
  #pragma unroll
  for (int t = 0; t < HEADS; ++t) {
    #pragma unroll
    for (int v = 0; v < 8; ++v) {
      long long r = row0 + v + mofs;
      if (r < nrows) h[r * F_OUT + t * F_PH + m] = acc[t][v];
    }
  }
}

// ---------------------------------------------------------------------------
// K2: per-node attention pre-dots  s_src[n,h] = h[n,h,:].a_src[h],  s_dst alike
// ---------------------------------------------------------------------------
__global__ void gat_node_scores(const float* __restrict__ h,
                                const float* __restrict__ a_src,
                                const float* __restrict__ a_dst,
                                float* __restrict__ s_src,
                                float* __restrict__ s_dst, int nNH) {
  int idx = blockIdx.x * blockDim.x + threadIdx.x;
  if (idx >= nNH) return;
  int n = idx >> 3, hd = idx & 7;
  const float* hp = h + (long long)n * F_OUT + hd * F_PH;
  const float* as = a_src + hd * F_PH;
  const float* ad = a_dst + hd * F_PH;
  float ss = 0.f, sd = 0.f;
  #pragma unroll
  for (int o = 0; o < F_PH; ++o) {
    float v = hp[o];
    ss += v * as[o];
    sd += v * ad[o];
  }
  s_src[idx] = ss;
  s_dst[idx] = sd;
}

// float atomic max via integer-ordering trick (works with -inf init)
__device__ __forceinline__ void atomicMaxF(float* addr, float val) {
  if (val >= 0.0f) atomicMax((int*)addr, __float_as_int(val));
  else             atomicMin((unsigned int*)addr, __float_as_uint(val));
}

// ---------------------------------------------------------------------------
// K3: edge logits + segment max.  e = leaky_relu(s_src[row] + s_dst[col])
// ---------------------------------------------------------------------------
__global__ void gat_edge_logits(const int* __restrict__ ei, int E,
                                const float* __restrict__ s_src,
                                const float* __restrict__ s_dst,
                                float* __restrict__ elog,
                                float* __restrict__ emax, int nEH) {
  int idx = blockIdx.x * blockDim.x + threadIdx.x;
  if (idx >= nEH) return;
  int e = idx >> 3, hd = idx & 7;
  int r = ei[e];        // destination (segment)
  int c = ei[E + e];    // source
  float v = s_src[r * HEADS + hd] + s_dst[c * HEADS + hd];
  v = (v >= 0.0f) ? v : NEG_SLOPE * v;
  elog[idx] = v;
  atomicMaxF(&emax[r * HEADS + hd], v);
}

// ---------------------------------------------------------------------------
// K4: ex = exp(e - emax[row]); denom[row] += ex
// ---------------------------------------------------------------------------
__global__ void gat_edge_exp(const int* __restrict__ ei, int E,
                             const float* __restrict__ emax,
                             float* __restrict__ elog,
                             float* __restrict__ denom, int nEH) {
  int idx = blockIdx.x * blockDim.x + threadIdx.x;
  if (idx >= nEH) return;
  int e = idx >> 3, hd = idx & 7;
  int r = ei[e];
  float ex = expf(elog[idx] - emax[r * HEADS + hd]);
  elog[idx] = ex;
  atomicAdd(&denom[r * HEADS + hd], ex);
}

// ---------------------------------------------------------------------------
// K5: out[row] += (ex/denom[row]) * h[col]
// 32 lanes per edge, each lane handles 4 contiguous floats (float4 load,
// 4 global_atomic_add_f32) -> 512B coalesced load + 512B contiguous atomics.
// ---------------------------------------------------------------------------
__global__ void gat_aggregate(const int* __restrict__ ei, int E,
                              const float* __restrict__ h,
                              const float* __restrict__ elog,
                              const float* __restrict__ denom,
                              float* __restrict__ out, int nWork) {
  int idx = blockIdx.x * blockDim.x + threadIdx.x;
  if (idx >= nWork) return;
  int e   = idx >> 5;
  int sub = idx & 31;
  int hd  = sub >> 2;
  int q   = sub & 3;
  int r = ei[e];
  int c = ei[E + e];
  float alpha = elog[e * HEADS + hd] / (denom[r * HEADS + hd] + 1e-16f);
  const float4 hv = *(const float4*)(h + (long long)c * F_OUT + hd * F_PH + q * 4);
  float* op = out + (long long)r * F_OUT + hd * F_PH + q * 4;
  atomicAdd(op + 0, alpha * hv.x);
  atomicAdd(op + 1, alpha * hv.y);
  atomicAdd(op + 2, alpha * hv.z);
  atomicAdd(op + 3, alpha * hv.w);
}

// ---------------------------------------------------------------------------
extern "C" void kernel_launch(void* const* d_in, const int* in_sizes, int n_in,
                              void* d_out, int out_size, void* d_ws, size_t ws_size,
                              hipStream_t stream) {
  const float* x     = (const float*)d_in[0];
  const int*   ei    = (const int*)d_in[1];
  const float* W     = (const float*)d_in[2];
  const float* a_src = (const float*)d_in[3];
  const float* a_dst = (const float*)d_in[4];
  float* out = (float*)d_out;

  const int N = in_sizes[0] / F_IN;
  const int E = in_sizes[1] / 2;

  // workspace carve-up
  float* h     = (float*)d_ws;                      // N * 128
  float* s_src = h     + (size_t)N * F_OUT;         // N * 8
  float* s_dst = s_src + (size_t)N * HEADS;         // N * 8
  float* emax  = s_dst + (size_t)N * HEADS;         // N * 8
  float* denom = emax  + (size_t)N * HEADS;         // N * 8
  float* elog  = denom + (size_t)N * HEADS;         // E * 8

  const int nOut  = N * F_OUT;
  const int nNH   = N * HEADS;
  const int nEH   = E * HEADS;
  const int nWork = E * 32;

  const int B = 256;

  // K0: init outputs / softmax accumulators
  {
    int total = nOut > nNH ? nOut : nNH;
    gat_init<<<(total + B - 1) / B, B, 0, stream>>>(out, emax, denom, nOut, nNH);
  }
  // K1: fp32 WMMA projection (8 waves/block, 128 rows per block)
  {
    int blocks = (N + 127) / 128;
    gat_proj_wmma<<<blocks, B, 0, stream>>>(x, W, h, N);
  }
  // K2: per-node scores
  gat_node_scores<<<(nNH + B - 1) / B, B, 0, stream>>>(h, a_src, a_dst, s_src, s_dst, nNH);
  // K3: edge logits + segment max
  gat_edge_logits<<<(nEH + B - 1) / B, B, 0, stream>>>(ei, E, s_src, s_dst, elog, emax, nEH);
  // K4: exp + denom
  gat_edge_exp<<<(nEH + B - 1) / B, B, 0, stream>>>(ei, E, emax, elog, denom, nEH);
  // K5: weighted scatter aggregation
  gat_aggregate<<<(nWork + B - 1) / B, B, 0, stream>>>(ei, E, h, elog, denom, out, nWork);
}